// BiRNN_71141838290985
// MI455X (gfx1250) — compile-verified
//
#include <hip/hip_runtime.h>

typedef __attribute__((ext_vector_type(2))) float v2f;
typedef __attribute__((ext_vector_type(8))) float v8f;

#define TT 128
#define BB 16
#define VOCAB 32000
#define HID 8
#define EMB 32
#define H2 16          // 2*HID
#define NTHREADS 1024
#define NWAVES 32
#define NTILE (VOCAB / 16)   // 2000

// ---------------------------------------------------------------------------
// Kernel 1: both RNN scans (sequential, tiny). One workgroup.
// threads 0..127: forward dir; 128..255: backward dir (uses b_x2 as both
// biases, faithful to the reference bug; backward states NOT re-reversed).
// Writes total_h [T][B][16] (fwd in [0:8], bwd in [8:16]) to workspace.
// ---------------------------------------------------------------------------
__global__ __launch_bounds__(256) void birnn_scan_kernel(
    const int* __restrict__ x,          // [T][B]
    const float* __restrict__ emb,      // [VOCAB][EMB]
    const float* __restrict__ W_x1, const float* __restrict__ b_x1,
    const float* __restrict__ W_h1, const float* __restrict__ b_h1,
    const float* __restrict__ W_x2, const float* __restrict__ b_x2,
    const float* __restrict__ W_h2,
    float* __restrict__ total_h)        // [T][B][H2]
{
    __shared__ float sWx[2][EMB][HID];
    __shared__ float sWh[2][HID][HID];
    __shared__ float sbx[2][HID];
    __shared__ float sbh[2][HID];
    __shared__ float sh[2][BB][HID];
    __shared__ float senc[2][BB][EMB];

    const int tid = threadIdx.x;
    const int dir = tid >> 7;      // 0 fwd, 1 bwd
    const int lt  = tid & 127;
    const int b   = lt >> 3;
    const int j   = lt & 7;

    for (int i = tid; i < 2 * EMB * HID; i += 256) {
        int d = i / (EMB * HID), r = i % (EMB * HID);
        sWx[d][r / HID][r % HID] = d ? W_x2[r] : W_x1[r];
    }
    for (int i = tid; i < 2 * HID * HID; i += 256) {
        int d = i / (HID * HID), r = i % (HID * HID);
        sWh[d][r / HID][r % HID] = d ? W_h2[r] : W_h1[r];
    }
    if (tid < 2 * HID) {
        int d = tid >> 3, r = tid & 7;
        sbx[d][r] = d ? b_x2[r] : b_x1[r];
        sbh[d][r] = d ? b_x2[r] : b_h1[r];   // reference bug: bwd hidden bias = b_x2
    }
    sh[dir][b][j] = 0.0f;                    // 256 threads cover all (dir,b,j)
    __syncthreads();

    for (int t = 0; t < TT; ++t) {
        const int ts = dir ? (TT - 1 - t) : t;
        // stage embeddings[x[ts]] : 16x32 floats per dir, 4 per thread
        #pragma unroll
        for (int i = 0; i < 4; ++i) {
            int e  = lt * 4 + i;
            int bb = e >> 5, ee = e & 31;
            int row = x[ts * BB + bb];
            senc[dir][bb][ee] = emb[row * EMB + ee];
        }
        __syncthreads();   // enc ready; prev-step h writes visible

        // store PRE-update hidden state (reference stores h before updating)
        total_h[(t * BB + b) * H2 + dir * HID + j] = sh[dir][b][j];

        float z = sbx[dir][j] + sbh[dir][j];
        #pragma unroll
        for (int e = 0; e < EMB; ++e) z += senc[dir][b][e] * sWx[dir][e][j];
        #pragma unroll
        for (int k = 0; k < HID; ++k) z += sh[dir][b][k] * sWh[dir][k][j];
        float hn = 1.0f / (1.0f + __expf(-z));
        __syncthreads();   // all reads of sh/senc done
        sh[dir][b][j] = hn;
    }
}

// ---------------------------------------------------------------------------
// Kernel 2: per t: logits = total_h[t] (16x16) @ Wout (16x32000) via
// V_WMMA_F32_16X16X4_F32 (4 k-slices), online log-softmax reduction,
// then recompute + non-temporal streamed write of log-probs.
// One block per t; 32 waves split the 2000 vocab tiles.
// ---------------------------------------------------------------------------
__global__ __launch_bounds__(NTHREADS) void birnn_logits_lsm_kernel(
    const float* __restrict__ total_h,   // [T][B][16]
    const float* __restrict__ Wout,      // [16][VOCAB]
    float* __restrict__ out)             // [T][B][VOCAB]
{
    __shared__ float red_m[NWAVES][16];
    __shared__ float red_s[NWAVES][16];
    __shared__ float sL[16];

    const int t    = blockIdx.x;
    const int tid  = threadIdx.x;
    const int lane = tid & 31;
    const int wave = tid >> 5;
    const int half = lane >> 4;   // 0: lanes 0-15, 1: lanes 16-31
    const int l16  = lane & 15;

    // A-matrix (16x4 f32) VGPR layout: lanes 0-15 row M=l16 hold K=kc,kc+1;
    // lanes 16-31 hold K=kc+2,kc+3 (ISA 7.12.2).
    const float* TH = total_h + t * (BB * H2);
    v2f a[4];
    #pragma unroll
    for (int c = 0; c < 4; ++c) {
        int k0 = 4 * c + 2 * half;
        a[c].x = TH[l16 * H2 + k0];
        a[c].y = TH[l16 * H2 + k0 + 1];
    }

    // B-matrix (4x16 f32): vgpr0 = row K=kc (lanes 0-15) / K=kc+2 (16-31),
    // vgpr1 = K=kc+1 / K=kc+3; N = n0 + l16. Same half/K pattern as A.
    const float* Wb = Wout + (2 * half) * VOCAB + l16;

    float m[8], s[8];
    #pragma unroll
    for (int r = 0; r < 8; ++r) { m[r] = -__builtin_inff(); s[r] = 0.0f; }

    // ---- Pass 1: online max / sum-exp over this wave's vocab tiles ----
    for (int tile = wave; tile < NTILE; tile += NWAVES) {
        const int n0 = tile * 16;
        v8f c = {};
        #pragma unroll
        for (int kc = 0; kc < 4; ++kc) {
            v2f bv;
            bv.x = Wb[(4 * kc) * VOCAB + n0];
            bv.y = Wb[(4 * kc + 1) * VOCAB + n0];
            c = __builtin_amdgcn_wmma_f32_16x16x4_f32(
                    false, a[kc], false, bv, (short)0, c, false, false);
        }
        #pragma unroll
        for (int r = 0; r < 8; ++r) {
            float xv = c[r];
            float nm = fmaxf(m[r], xv);
            s[r] = s[r] * __expf(m[r] - nm) + __expf(xv - nm);
            m[r] = nm;
        }
    }

    // reduce across the 16 lanes of each half (they share row r / r+8)
    #pragma unroll
    for (int r = 0; r < 8; ++r) {
        #pragma unroll
        for (int mask = 1; mask < 16; mask <<= 1) {
            float om = __shfl_xor(m[r], mask, 32);
            float os = __shfl_xor(s[r], mask, 32);
            float nm = fmaxf(m[r], om);
            s[r] = s[r] * __expf(m[r] - nm) + os * __expf(om - nm);
            m[r] = nm;
        }
    }
    if (l16 == 0) {
        #pragma unroll
        for (int r = 0; r < 8; ++r) {
            red_m[wave][r + 8 * half] = m[r];
            red_s[wave][r + 8 * half] = s[r];
        }
    }
    __syncthreads();
    if (tid < 16) {
        float mm = red_m[0][tid], ss = red_s[0][tid];
        for (int w = 1; w < NWAVES; ++w) {
            float om = red_m[w][tid], os = red_s[w][tid];
            float nm = fmaxf(mm, om);
            ss = ss * __expf(mm - nm) + os * __expf(om - nm);
            mm = nm;
        }
        sL[tid] = mm + __logf(ss);   // log-sum-exp per batch row
    }
    __syncthreads();

    float L[8];
    #pragma unroll
    for (int r = 0; r < 8; ++r) L[r] = sL[r + 8 * half];

    float* orow = out + (size_t)t * (BB * VOCAB);

    // ---- Pass 2: recompute tiles (Wout is L2-resident) and stream out ----
    for (int tile = wave; tile < NTILE; tile += NWAVES) {
        const int n0 = tile * 16;
        v8f c = {};
        #pragma unroll
        for (int kc = 0; kc < 4; ++kc) {
            v2f bv;
            bv.x = Wb[(4 * kc) * VOCAB + n0];
            bv.y = Wb[(4 * kc + 1) * VOCAB + n0];
            c = __builtin_amdgcn_wmma_f32_16x16x4_f32(
                    false, a[kc], false, bv, (short)0, c, false, false);
        }
        const int n = n0 + l16;
        #pragma unroll
        for (int r = 0; r < 8; ++r) {
            const int row = r + 8 * half;    // C/D layout: vgpr r -> M=r (+8 hi half)
            __builtin_nontemporal_store(c[r] - L[r], &orow[row * VOCAB + n]);
        }
    }
}

extern "C" void kernel_launch(void* const* d_in, const int* in_sizes, int n_in,
                              void* d_out, int out_size, void* d_ws, size_t ws_size,
                              hipStream_t stream) {
    const int*   x    = (const int*)d_in[0];
    const float* emb  = (const float*)d_in[1];
    const float* W_x1 = (const float*)d_in[2];
    const float* b_x1 = (const float*)d_in[3];
    const float* W_h1 = (const float*)d_in[4];
    const float* b_h1 = (const float*)d_in[5];
    const float* W_x2 = (const float*)d_in[6];
    const float* b_x2 = (const float*)d_in[7];
    const float* W_h2 = (const float*)d_in[8];
    // d_in[9] = b_h2: unused by the reference (faithful bug)
    const float* Wout = (const float*)d_in[10];
    float* out     = (float*)d_out;
    float* total_h = (float*)d_ws;       // 128*16*16*4 = 128 KB scratch

    hipLaunchKernelGGL(birnn_scan_kernel, dim3(1), dim3(256), 0, stream,
                       x, emb, W_x1, b_x1, W_h1, b_h1, W_x2, b_x2, W_h2, total_h);
    hipLaunchKernelGGL(birnn_logits_lsm_kernel, dim3(TT), dim3(NTHREADS), 0, stream,
                       total_h, Wout, out);
}